// Encoder_12446815224230
// MI455X (gfx1250) — compile-verified
//
#include <hip/hip_runtime.h>

// ---------------------------------------------------------------------------
// GCN encoder for MI455X (gfx1250, wave32).
// Dense GEMMs use V_WMMA_F32_16X16X4_F32 (fp32 A/B/C -> full precision).
// Message passing uses hardware global FP32 atomics.
// ---------------------------------------------------------------------------

typedef __attribute__((ext_vector_type(2))) float v2f;
typedef __attribute__((ext_vector_type(8))) float v8f;

#define LATENT 64
#define BN_EPS 1e-5f

// ---------------------------------------------------------------------------
// Generic fp32 WMMA GEMM: C = A(MxK) @ B(KxN) [+bias][+relu / fc2-split]
// One wave (32 threads) computes one 16x16 output tile; K stepped by 4.
// Requires M%16==0, N%16==0, K%4==0 (true for all shapes here).
// mode 0: plain store.  mode 1: +bias, relu.  mode 2: +bias, split mu/logsig.
// ---------------------------------------------------------------------------
__global__ __launch_bounds__(32) void wmma_gemm_f32(
    const float* __restrict__ A, const float* __restrict__ B,
    const float* __restrict__ bias, float* __restrict__ C,
    int M, int N, int K, int mode)
{
  const int ntiles = N >> 4;
  const int tm = blockIdx.x / ntiles;
  const int tn = blockIdx.x % ntiles;
  const int lane = threadIdx.x;       // 0..31
  const int half = lane >> 4;         // 0 or 1
  const int l16  = lane & 15;

  // A fragment: lane -> row M = tm*16 + l16 ; VGPR v -> K = k + v + 2*half
  // B fragment: lane -> col N = tn*16 + l16 ; VGPR v -> K = k + v + 2*half
  const long long mrow = (long long)tm * 16 + l16;
  const int ncol = tn * 16 + l16;
  const float* __restrict__ arow = A + mrow * (long long)K;

  v8f acc = {0.f,0.f,0.f,0.f,0.f,0.f,0.f,0.f};
  for (int k = 0; k < K; k += 4) {
    const int ka = k + 2 * half;                 // even -> 8B aligned
    v2f a = *(const v2f*)(arow + ka);            // A[m][ka], A[m][ka+1]
    v2f b;
    b[0] = B[(long long)ka * N + ncol];          // B[ka  ][n]
    b[1] = B[(long long)(ka + 1) * N + ncol];    // B[ka+1][n]
    acc = __builtin_amdgcn_wmma_f32_16x16x4_f32(
        /*neg_a=*/false, a, /*neg_b=*/false, b,
        /*c_mod=*/(short)0, acc, /*reuse_a=*/false, /*reuse_b=*/false);
  }

  // D layout: VGPR r -> row M = tm*16 + r + 8*half ; lane -> col N = tn*16+l16
  for (int r = 0; r < 8; ++r) {
    const long long row = (long long)tm * 16 + r + 8 * half;
    const int col = tn * 16 + l16;
    float v = acc[r];
    if (mode != 0) v += bias[col];
    if (mode == 1) v = fmaxf(v, 0.0f);
    if (mode == 2) {
      // N == 2*LATENT; first half -> mu, second half -> log_sigma
      if (col < LATENT) C[row * LATENT + col] = v;
      else C[(long long)M * LATENT + row * LATENT + (col - LATENT)] = v;
    } else {
      C[row * (long long)N + col] = v;
    }
  }
}

// ---------------------------------------------------------------------------
// Degree / normalization
// ---------------------------------------------------------------------------
__global__ __launch_bounds__(256) void deg_init(float* __restrict__ deg, int n) {
  int i = blockIdx.x * 256 + threadIdx.x;
  if (i < n) deg[i] = 1.0f;                      // self-loop contribution
}
__global__ __launch_bounds__(256) void deg_accum(const long long* __restrict__ dst,
                                                 long long nE, float* __restrict__ deg) {
  long long e = (long long)blockIdx.x * 256 + threadIdx.x;
  if (e < nE) atomicAdd(&deg[dst[e]], 1.0f);
}
__global__ __launch_bounds__(256) void deg_rsqrt(float* __restrict__ deg, int n) {
  int i = blockIdx.x * 256 + threadIdx.x;
  if (i < n) deg[i] = rsqrtf(deg[i]);            // deg >= 1 always (self loops)
}

// ---------------------------------------------------------------------------
// agg[i][f] = h[i][f] * isq[i]^2 + bias[f]   (self-loop edge + conv bias)
// ---------------------------------------------------------------------------
__global__ __launch_bounds__(256) void agg_init(
    const float* __restrict__ h, const float* __restrict__ isq,
    const float* __restrict__ bias, float* __restrict__ agg,
    long long total, int logF)
{
  long long idx = (long long)blockIdx.x * 256 + threadIdx.x;
  if (idx >= total) return;
  long long i = idx >> logF;
  int f = (int)(idx & ((1 << logF) - 1));
  float w = isq[i];
  agg[idx] = h[idx] * w * w + bias[f];
}

// ---------------------------------------------------------------------------
// agg[dst][f] += h[src][f] * isq[src]*isq[dst]  over all real edges
// ---------------------------------------------------------------------------
__global__ __launch_bounds__(256) void edge_scatter(
    const long long* __restrict__ src, const long long* __restrict__ dst,
    long long nE, const float* __restrict__ h, const float* __restrict__ isq,
    float* __restrict__ agg, int logF)
{
  const int F = 1 << logF;
  long long idx = (long long)blockIdx.x * 256 + threadIdx.x;
  long long total = nE << logF;
  if (idx >= total) return;
  long long e = idx >> logF;
  int f = (int)(idx & (F - 1));
  long long s = src[e];
  long long d = dst[e];
  float norm = isq[s] * isq[d];
  atomicAdd(&agg[d * F + f], h[s * F + f] * norm);
}

// ---------------------------------------------------------------------------
// BatchNorm over 32 features (double-precision accumulation for accuracy)
// ---------------------------------------------------------------------------
__global__ __launch_bounds__(64) void stats_zero(double* __restrict__ stats) {
  stats[threadIdx.x] = 0.0;
}
__global__ __launch_bounds__(256) void bn_stats(
    const float* __restrict__ h, int n, double* __restrict__ stats)
{
  const int tid = threadIdx.x;
  const int f = tid & 31;
  const int grp = tid >> 5;                       // 8 row-groups per block
  double s = 0.0, s2 = 0.0;
  for (int i = blockIdx.x * 8 + grp; i < n; i += gridDim.x * 8) {
    float v = h[(long long)i * 32 + f];
    s += v; s2 += (double)v * v;
  }
  __shared__ double sh[256], sh2[256];
  sh[tid] = s; sh2[tid] = s2; __syncthreads();
  if (tid < 128) { sh[tid] += sh[tid + 128]; sh2[tid] += sh2[tid + 128]; } __syncthreads();
  if (tid <  64) { sh[tid] += sh[tid +  64]; sh2[tid] += sh2[tid +  64]; } __syncthreads();
  if (tid <  32) {
    atomicAdd(&stats[f],      sh[tid]  + sh[tid + 32]);
    atomicAdd(&stats[32 + f], sh2[tid] + sh2[tid + 32]);
  }
}
__global__ __launch_bounds__(32) void bn_finalize(
    const double* __restrict__ stats, const float* __restrict__ gamma,
    const float* __restrict__ beta, double inv_n, float* __restrict__ scsh)
{
  int f = threadIdx.x;
  double mean = stats[f] * inv_n;
  double var  = stats[32 + f] * inv_n - mean * mean;
  float scale = gamma[f] * rsqrtf((float)var + BN_EPS);
  scsh[f]      = scale;
  scsh[32 + f] = beta[f] - (float)mean * scale;
}
__global__ __launch_bounds__(256) void bn_apply_relu(
    float* __restrict__ h, const float* __restrict__ scsh, long long total)
{
  long long idx = (long long)blockIdx.x * 256 + threadIdx.x;
  if (idx >= total) return;
  int f = (int)(idx & 31);
  h[idx] = fmaxf(h[idx] * scsh[f] + scsh[32 + f], 0.0f);
}

// ---------------------------------------------------------------------------
// Graph mean pooling
// ---------------------------------------------------------------------------
__global__ __launch_bounds__(256) void zero_f32(float* __restrict__ p, long long total) {
  long long idx = (long long)blockIdx.x * 256 + threadIdx.x;
  if (idx < total) p[idx] = 0.0f;
}
__global__ __launch_bounds__(256) void pool_accum(
    const float* __restrict__ h, const long long* __restrict__ batch,
    long long total, int logF, float* __restrict__ gsum, float* __restrict__ gcnt)
{
  long long idx = (long long)blockIdx.x * 256 + threadIdx.x;
  if (idx >= total) return;
  long long i = idx >> logF;
  int f = (int)(idx & ((1 << logF) - 1));
  long long g = batch[i];
  atomicAdd(&gsum[(g << logF) + f], h[idx]);
  if (f == 0) atomicAdd(&gcnt[g], 1.0f);
}
__global__ __launch_bounds__(256) void pool_div(
    float* __restrict__ gsum, const float* __restrict__ gcnt,
    long long total, int logF)
{
  long long idx = (long long)blockIdx.x * 256 + threadIdx.x;
  if (idx >= total) return;
  gsum[idx] /= fmaxf(gcnt[idx >> logF], 1.0f);
}

// ---------------------------------------------------------------------------
// Host-side orchestration
// ---------------------------------------------------------------------------
static inline unsigned blocks_for(long long total, int bs) {
  return (unsigned)((total + bs - 1) / bs);
}

extern "C" void kernel_launch(void* const* d_in, const int* in_sizes, int n_in,
                              void* d_out, int out_size, void* d_ws, size_t ws_size,
                              hipStream_t stream)
{
  const float*      x     = (const float*)d_in[0];
  const long long*  edge  = (const long long*)d_in[1];   // int64: [2, nE]
  const long long*  batch = (const long long*)d_in[2];   // int64: [n]
  const float*      W1    = (const float*)d_in[3];
  const float*      b1    = (const float*)d_in[4];
  const float*      gamma = (const float*)d_in[5];
  const float*      beta  = (const float*)d_in[6];
  const float*      W2    = (const float*)d_in[7];
  const float*      b2    = (const float*)d_in[8];
  const float*      Wfc1  = (const float*)d_in[9];
  const float*      bfc1  = (const float*)d_in[10];
  const float*      Wfc2  = (const float*)d_in[11];
  const float*      bfc2  = (const float*)d_in[12];
  float*            out   = (float*)d_out;

  const int       F1   = in_sizes[4];                    // 32
  const int       n    = in_sizes[0] / F1;               // 150000
  const long long nE   = (long long)in_sizes[1] / 2;     // 2400000
  const int       F2   = in_sizes[8];                    // 64
  const int       FC1N = in_sizes[10];                   // 128
  const int       FC2N = in_sizes[12];                   // 128 == 2*LATENT
  const int       nG   = out_size / (2 * LATENT);        // 8192

  const long long* srcI = edge;        // edge_index[0]
  const long long* dstI = edge + nE;   // edge_index[1]

  // Bump-allocate workspace (256B aligned); total ~122 MB.
  char* wsb = (char*)d_ws;
  size_t off = 0;
  auto walloc = [&](size_t bytes) -> void* {
    void* p = wsb + off;
    off += (bytes + 255) & ~(size_t)255;
    return p;
  };
  double* stats = (double*)walloc(64 * sizeof(double));          // sum[32], sumsq[32]
  float*  scsh  = (float*) walloc(64 * sizeof(float));           // scale[32], shift[32]
  float*  isq   = (float*) walloc((size_t)n * sizeof(float));
  float*  h1    = (float*) walloc((size_t)n * F1 * sizeof(float));
  float*  agg1  = (float*) walloc((size_t)n * F1 * sizeof(float));
  float*  h2    = (float*) walloc((size_t)n * F2 * sizeof(float));
  float*  agg2  = (float*) walloc((size_t)n * F2 * sizeof(float));
  float*  gsum  = (float*) walloc((size_t)nG * F2 * sizeof(float));
  float*  gcnt  = (float*) walloc((size_t)nG * sizeof(float));
  float*  gfc1  = (float*) walloc((size_t)nG * FC1N * sizeof(float));
  (void)ws_size;

  // 1) degree -> deg^{-1/2}   (self-loops included via init to 1.0)
  deg_init  <<<blocks_for(n, 256), 256, 0, stream>>>(isq, n);
  deg_accum <<<blocks_for(nE, 256), 256, 0, stream>>>(dstI, nE, isq);
  deg_rsqrt <<<blocks_for(n, 256), 256, 0, stream>>>(isq, n);

  // 2) conv1: h1 = x @ W1 (WMMA), then normalized scatter-add + bias
  wmma_gemm_f32<<<(n / 16) * (F1 / 16), 32, 0, stream>>>(x, W1, nullptr, h1, n, F1, F1, 0);
  agg_init    <<<blocks_for((long long)n * F1, 256), 256, 0, stream>>>(h1, isq, b1, agg1, (long long)n * F1, 5);
  edge_scatter<<<blocks_for(nE * F1, 256), 256, 0, stream>>>(srcI, dstI, nE, h1, isq, agg1, 5);

  // 3) BatchNorm(32 feat) + ReLU, fused scale/shift, in place on agg1
  stats_zero   <<<1, 64, 0, stream>>>(stats);
  bn_stats     <<<512, 256, 0, stream>>>(agg1, n, stats);
  bn_finalize  <<<1, 32, 0, stream>>>(stats, gamma, beta, 1.0 / (double)n, scsh);
  bn_apply_relu<<<blocks_for((long long)n * F1, 256), 256, 0, stream>>>(agg1, scsh, (long long)n * F1);

  // 4) conv2: h2 = agg1 @ W2 (WMMA), normalized scatter-add + bias
  wmma_gemm_f32<<<(n / 16) * (F2 / 16), 32, 0, stream>>>(agg1, W2, nullptr, h2, n, F2, F1, 0);
  agg_init    <<<blocks_for((long long)n * F2, 256), 256, 0, stream>>>(h2, isq, b2, agg2, (long long)n * F2, 6);
  edge_scatter<<<blocks_for(nE * F2, 256), 256, 0, stream>>>(srcI, dstI, nE, h2, isq, agg2, 6);

  // 5) graph mean pooling into gsum (8192 x 64)
  zero_f32  <<<blocks_for((long long)nG * F2, 256), 256, 0, stream>>>(gsum, (long long)nG * F2);
  zero_f32  <<<blocks_for((long long)nG, 256), 256, 0, stream>>>(gcnt, (long long)nG);
  pool_accum<<<blocks_for((long long)n * F2, 256), 256, 0, stream>>>(agg2, batch, (long long)n * F2, 6, gsum, gcnt);
  pool_div  <<<blocks_for((long long)nG * F2, 256), 256, 0, stream>>>(gsum, gcnt, (long long)nG * F2, 6);

  // 6) FC1 (+bias, relu) and FC2 (+bias, split mu/log_sigma) via WMMA
  wmma_gemm_f32<<<(nG / 16) * (FC1N / 16), 32, 0, stream>>>(gsum, Wfc1, bfc1, gfc1, nG, FC1N, F2, 1);
  wmma_gemm_f32<<<(nG / 16) * (FC2N / 16), 32, 0, stream>>>(gfc1, Wfc2, bfc2, out, nG, FC2N, FC1N, 2);
}